// DRMM_51548197486764
// MI455X (gfx1250) — compile-verified
//
#include <hip/hip_runtime.h>
#include <math.h>

// Problem constants (fixed by the reference).
#define QL     50      // query terms
#define EMB    50      // embedding dim (K of the GEMM)
#define DL     2000    // doc terms
#define NBINS  11
#define QPAD   64      // Q padded to 4 tiles of 16
#define EPAD   52      // K row stride (WMMA reads reach col 51); 52r%64 distinct for 16 rows
#define CHUNK  64      // d columns per chunk
#define NCHUNK 32      // 31*64 + 16 = 2000
#define SIMP   84      // sim row stride: 8*84 % 64 == 32 -> disjoint bank halves

typedef float v2f __attribute__((ext_vector_type(2)));
typedef float v8f __attribute__((ext_vector_type(8)));

#if defined(__gfx1250__) && __has_builtin(__builtin_amdgcn_wmma_f32_16x16x4_f32)
#define USE_WMMA_F32 1
#else
#define USE_WMMA_F32 0
#endif

#if defined(__gfx1250__)
#define USE_ASYNC 1
#else
#define USE_ASYNC 0
#endif

// Stage one 64-term d-chunk (rem rows x 50 floats) into LDS.
// Async path: GLOBAL_LOAD_ASYNC_TO_LDS_B64 (ASYNCcnt), bypassing VGPRs.
// Rows are 200B and pairs 8B -> every transfer is 8-byte aligned on both sides.
__device__ __forceinline__ void issue_chunk(const float* __restrict__ db, int dbase,
                                            int rem, float* dnp, int tid) {
#if USE_ASYNC
    const int npair = rem * (EMB / 2);          // 25 float2 per row
    for (int i = tid; i < npair; i += 256) {
        const int row = i / 25;
        const int pr  = i - row * 25;
        // low 32 bits of a flat shared pointer == LDS byte address
        unsigned ldsa = (unsigned)(size_t)(&dnp[row * EPAD + pr * 2]);
        unsigned goff = (unsigned)((dbase + row) * EMB + pr * 2) * 4u;
        asm volatile("global_load_async_to_lds_b64 %0, %1, %2"
                     :: "v"(ldsa), "v"(goff), "s"(db)
                     : "memory");
    }
#else
    const int nel = rem * EMB;
    for (int i = tid; i < nel; i += 256)
        dnp[(i / EMB) * EPAD + (i % EMB)] = db[(size_t)dbase * EMB + i];
#endif
}

__device__ __forceinline__ void wait_async() {
#if USE_ASYNC
    asm volatile("s_wait_asynccnt 0x0" ::: "memory");
#endif
}

__global__ __launch_bounds__(256)
void drmm_batch_kernel(const float* __restrict__ qemb,
                       const float* __restrict__ demb,
                       const float* __restrict__ W1,
                       const float* __restrict__ b1,
                       const float* __restrict__ Wg,
                       const int*   __restrict__ qids,
                       const int*   __restrict__ dids,
                       float* __restrict__ gate_ws,
                       float* __restrict__ ffn_ws)
{
    __shared__ float qn[QPAD * EPAD];        // normalized query embeddings
    __shared__ float dn[2][CHUNK * EPAD];    // double-buffered doc chunks
    __shared__ float sm[QPAD * SIMP];        // sim tile (64 q x 64 d)
    __shared__ float hist[QPAD * NBINS];     // per-q histogram counts
    __shared__ int   qval[QPAD];
    __shared__ int   dval[2][CHUNK];
    __shared__ float wred[8];

    const int b    = blockIdx.x;
    const int tid  = threadIdx.x;
    const int lane = tid & 31;
    const int wid  = tid >> 5;
    const int hi   = lane >> 4;   // 0: lanes 0-15, 1: lanes 16-31
    const int lo   = lane & 15;

    const float* qb = qemb + (size_t)b * QL * EMB;
    const float* db = demb + (size_t)b * DL * EMB;

    // ---- zero LDS (pad rows/cols must be zero for WMMA K-padding) ----
    for (int i = tid; i < QPAD * EPAD;      i += 256) qn[i]    = 0.f;
    for (int i = tid; i < 2 * CHUNK * EPAD; i += 256) dn[0][i] = 0.f;
    for (int i = tid; i < QPAD * NBINS;     i += 256) hist[i]  = 0.f;
    __syncthreads();   // zeroing committed before async writes may land

    // ---- kick off the async stream of chunk 0 while we set up q ----
    issue_chunk(db, 0, CHUNK, dn[0], tid);

    // ---- load q embeddings; fuse the term-gating dot (raw q, pre-norm) ----
    float gpart = 0.f;
    for (int i = tid; i < QL * EMB; i += 256) {
        float v = qb[i];
        qn[(i / EMB) * EPAD + (i % EMB)] = v;
        gpart += v * Wg[i];
    }
    #pragma unroll
    for (int off = 16; off > 0; off >>= 1) gpart += __shfl_down(gpart, off, 32);
    if (lane == 0) wred[wid] = gpart;
    if (tid < QPAD) qval[tid] = (tid < QL) ? (qids[b * QL + tid] > 0) : 0;
    __syncthreads();
    if (tid == 0) {
        float g = 0.f;
        #pragma unroll
        for (int w = 0; w < 8; ++w) g += wred[w];
        gate_ws[b] = g;
    }

    // ---- normalize q rows: q / (||q|| + 1e-8) ----
    if (tid < QL) {
        float ss = 0.f;
        for (int k = 0; k < EMB; ++k) { float v = qn[tid * EPAD + k]; ss += v * v; }
        float inv = 1.f / (sqrtf(ss) + 1e-8f);
        for (int k = 0; k < EMB; ++k) qn[tid * EPAD + k] *= inv;
    }
    __syncthreads();

    // ---- preload A fragments: per-wave fixed q tile, resident in VGPRs.
    //      ISA 32-bit A 16x4 layout: lane (M=lane&15), VGPR v: K = 4j + 2*(lane>>4) + v.
    const int qt = wid >> 1;   // waves {0,1}->qtile 0 ... {6,7}->qtile 3
    v2f aK[13];
    #pragma unroll
    for (int j = 0; j < 13; ++j) {
        const int col = 4 * j + 2 * hi;
        aK[j].x = qn[(qt * 16 + lo) * EPAD + col];
        aK[j].y = qn[(qt * 16 + lo) * EPAD + col + 1];
    }

    // ---- stream doc terms: double-buffered async pipeline ----
    for (int c = 0; c < NCHUNK; ++c) {
        const int p     = c & 1;
        const int dbase = c * CHUNK;
        const int rem   = min(CHUNK, DL - dbase);

        if (tid < CHUNK)
            dval[p][tid] = (tid < rem) ? (dids[b * DL + dbase + tid] > 0) : 0;

        wait_async();        // this wave's chunk-c transfers have landed
        __syncthreads();     // ... and everyone else's too

        // overlap: stream chunk c+1 into the other buffer during compute
        if (c + 1 < NCHUNK) {
            const int nbase = dbase + CHUNK;
            issue_chunk(db, nbase, min(CHUNK, DL - nbase), dn[p ^ 1], tid);
        }

        // normalize chunk rows
        if (tid < rem) {
            float ss = 0.f;
            for (int k = 0; k < EMB; ++k) { float v = dn[p][tid * EPAD + k]; ss += v * v; }
            float inv = 1.f / (sqrtf(ss) + 1e-8f);
            for (int k = 0; k < EMB; ++k) dn[p][tid * EPAD + k] *= inv;
        }
        __syncthreads();

        // WMMA: each wave computes two 16x16 output tiles (4 qtiles x 4 dtiles)
        #pragma unroll
        for (int i2 = 0; i2 < 2; ++i2) {
            const int dt = (wid & 1) * 2 + i2;
            if (dt * 16 < rem) {
                v8f acc = {};
#if USE_WMMA_F32
                #pragma unroll
                for (int j = 0; j < 13; ++j) {
                    const int col = 4 * j + 2 * hi;
                    v2f bK;
                    bK.x = dn[p][(dt * 16 + lo) * EPAD + col];
                    bK.y = dn[p][(dt * 16 + lo) * EPAD + col + 1];
                    acc = __builtin_amdgcn_wmma_f32_16x16x4_f32(
                        /*neg_a=*/false, aK[j], /*neg_b=*/false, bK,
                        /*c_mod=*/(short)0, acc, /*reuse_a=*/false, /*reuse_b=*/false);
                }
#else
                // exact scalar fallback (keeps compile feedback alive)
                #pragma unroll
                for (int i = 0; i < 8; ++i) {
                    const int row = qt * 16 + i + 8 * hi;
                    const int col = dt * 16 + lo;
                    float s = 0.f;
                    for (int k = 0; k < EMB; ++k)
                        s += qn[row * EPAD + k] * dn[p][col * EPAD + k];
                    acc[i] = s;
                }
#endif
                // C/D layout: VGPR i -> row M = i + 8*hi, col N = lane&15
                #pragma unroll
                for (int i = 0; i < 8; ++i)
                    sm[(qt * 16 + i + 8 * hi) * SIMP + dt * 16 + lo] = acc[i];
            }
        }
        __syncthreads();

        // histogram: wave w owns q rows {w, w+8, ...}; counts via ballots
        // (no LDS atomics, wave-uniform scalar accumulation)
        for (int qrow = wid; qrow < QPAD; qrow += 8) {
            if (!qval[qrow]) continue;   // uniform per wave
            unsigned cnt[NBINS] = {0,0,0,0,0,0,0,0,0,0,0};
            #pragma unroll
            for (int it = 0; it < CHUNK / 32; ++it) {
                const int j = it * 32 + lane;
                const float s = sm[qrow * SIMP + j];
                const unsigned long long vmask = __ballot(dval[p][j] != 0);
                int bin = (int)((s + 1.000001f) * 5.0f);   // 0.5*(NBINS-1)
                bin = bin < 0 ? 0 : (bin > 10 ? 10 : bin);
                #pragma unroll
                for (int bb = 0; bb < NBINS; ++bb)
                    cnt[bb] += (unsigned)__popcll(__ballot(bin == bb) & vmask);
            }
            if (lane == 0) {
                #pragma unroll
                for (int bb = 0; bb < NBINS; ++bb)
                    hist[qrow * NBINS + bb] += (float)cnt[bb];
            }
        }
        __syncthreads();   // protect sm/dval before next iteration's writes
    }

    // ---- out_ffn[b] = dot(log(hist + 1e-5), W1) + b1 ----
    float part = 0.f;
    for (int i = tid; i < QL * NBINS; i += 256)
        part += logf(hist[i] + 1e-5f) * W1[i];
    #pragma unroll
    for (int off = 16; off > 0; off >>= 1) part += __shfl_down(part, off, 32);
    if (lane == 0) wred[wid] = part;
    __syncthreads();
    if (tid == 0) {
        float t = 0.f;
        #pragma unroll
        for (int w = 0; w < 8; ++w) t += wred[w];
        ffn_ws[b] = t + b1[0];
    }
}

// Softmax over the batch + faithful (B,1)*(B,) broadcast-sum:
// score[i] = out_ffn[i] * sum_j softmax(gate)[j]
__global__ __launch_bounds__(256)
void drmm_score_kernel(const float* __restrict__ gate_ws,
                       const float* __restrict__ ffn_ws,
                       float* __restrict__ out)
{
    __shared__ float s[256];
    const int t = threadIdx.x;
    const float g = gate_ws[t];
    s[t] = g;
    __syncthreads();
    for (int off = 128; off > 0; off >>= 1) {
        if (t < off) s[t] = fmaxf(s[t], s[t + off]);
        __syncthreads();
    }
    const float m = s[0];
    __syncthreads();
    const float e = expf(g - m);
    s[t] = e;
    __syncthreads();
    for (int off = 128; off > 0; off >>= 1) {
        if (t < off) s[t] = s[t] + s[t + off];
        __syncthreads();
    }
    const float denom = s[0];
    __syncthreads();
    s[t] = e / denom;          // softmax value
    __syncthreads();
    for (int off = 128; off > 0; off >>= 1) {
        if (t < off) s[t] = s[t] + s[t + off];
        __syncthreads();
    }
    out[t] = ffn_ws[t] * s[0];
}

extern "C" void kernel_launch(void* const* d_in, const int* in_sizes, int n_in,
                              void* d_out, int out_size, void* d_ws, size_t ws_size,
                              hipStream_t stream) {
    (void)n_in; (void)ws_size; (void)out_size;
    const float* qemb = (const float*)d_in[0];
    const float* demb = (const float*)d_in[1];
    const float* W1   = (const float*)d_in[2];
    const float* b1   = (const float*)d_in[3];
    const float* Wg   = (const float*)d_in[4];
    const int*   qids = (const int*)d_in[5];
    const int*   dids = (const int*)d_in[6];

    const int B = in_sizes[0] / (QL * EMB);   // 256
    float* gate_ws = (float*)d_ws;
    float* ffn_ws  = gate_ws + B;
    float* outp    = (float*)d_out;

    drmm_batch_kernel<<<B, 256, 0, stream>>>(qemb, demb, W1, b1, Wg, qids, dids,
                                             gate_ws, ffn_ws);
    drmm_score_kernel<<<1, B, 0, stream>>>(gate_ws, ffn_ws, outp);
}